// FrontierExplorationPolicy_38190849196751
// MI455X (gfx1250) — compile-verified
//
#include <hip/hip_runtime.h>
#include <stdint.h>

// ---------------------------------------------------------------------------
// Frontier exploration policy: goal = dilate(erode(frontier, disk10), disk1)
//                                     - erode(frontier, disk10)
// Bandwidth-bound (59 MB min traffic @ 23.3 TB/s ~ 2.5 us). Strategy:
//   1) bit-pack frontier via wave32 ballot  (29.5 MB -> 0.9 MB)
//   2) TDM (tensor_load_to_lds) stages packed tiles into LDS  [CDNA5 path]
//   3) disk erosion = AND of shifted bitmasks (7 distinct row widths)
//   4) cross dilation, border extraction, bit->f32 coalesced stores
// ---------------------------------------------------------------------------

typedef unsigned int v4u   __attribute__((ext_vector_type(4)));
typedef int          v8i_t __attribute__((ext_vector_type(8)));
typedef int          v4i_t __attribute__((ext_vector_type(4)));

#define MAPW  960
#define WW    30            // 960 / 32 packed words per row
#define BAND  16            // output rows per block
#define ROWS  (BAND + 22)   // 38: band + 11-row halo each side (erode 10 + dilate 1)
#define ERN   (BAND + 2)    // eroded rows needed (band + dilation halo)
#define PLANE ((size_t)MAPW * (size_t)MAPW)
#define NCH   24
#define EXPLORED 1

// ---------------------------------------------------------------------------
// Kernel 1: frontier = (explored == 0), packed 32 pixels/word via wave32 ballot.
// Batch on blockIdx.y so per-thread addressing is a single fused multiply-add
// (no 64-bit magic-number division in the 29.5 MB-read hot path).
// ---------------------------------------------------------------------------
__global__ __launch_bounds__(256) void fep_pack(const float* __restrict__ mf,
                                                unsigned* __restrict__ bits) {
  const unsigned pix = blockIdx.x * 256u + threadIdx.x;   // pixel within plane
  const unsigned b   = blockIdx.y;                        // batch
  const float* src = mf + ((size_t)b * NCH + EXPLORED) * PLANE;
  float v = src[pix];
  unsigned m = __builtin_amdgcn_ballot_w32(v == 0.0f);    // wave32 ballot
  if ((threadIdx.x & 31u) == 0u)
    bits[(((size_t)b * PLANE) + pix) >> 5] = m;
}

// ---------------------------------------------------------------------------
// Kernel 2: morphology on packed bits. One block = one 16-row band of a batch.
// ---------------------------------------------------------------------------
__global__ __launch_bounds__(256) void fep_morph(const unsigned* __restrict__ bits,
                                                 float* __restrict__ out) {
  __shared__ unsigned sF[ROWS][WW];        // packed frontier (band + halo)
  __shared__ unsigned sE[7][ROWS][WW];     // horizontal erosions, 7 width classes
  __shared__ unsigned sEr[ERN][WW];        // disk-eroded rows y0-1 .. y0+16
  __shared__ unsigned sG[BAND][WW];        // goal bits

  const int tid = threadIdx.x;
  const int b   = blockIdx.y;
  const int y0  = blockIdx.x * BAND;
  const int gr0 = y0 - 11;                 // global row of sF[0]
  int r0 = gr0 < 0 ? 0 : gr0;              // first valid global row to DMA
  int r1 = gr0 + ROWS - 1; if (r1 > MAPW - 1) r1 = MAPW - 1;
  const int nrows = r1 - r0 + 1;

  // Zero halo rows that fall outside the image (zero-padding semantics).
  for (int i = tid; i < ROWS * WW; i += 256) {
    int rr = i / WW;
    int gr = gr0 + rr;
    if (gr < 0 || gr >= MAPW) sF[rr][i % WW] = 0u;
  }

  // --- CDNA5 Tensor Data Mover: DMA the packed tile (nrows x 30 u32) to LDS.
  // 2-D descriptor: data_size=4B, tensor_dim0=30, stride0=30, tensor_dim1=960,
  // tile = 30 x nrows, global_addr = tile start, type=2 ("image").
  if (tid < 32) {   // one wave issues the TDM op (EXEC ignored by TDM)
    unsigned long long ga =
        (unsigned long long)(uintptr_t)bits +
        (((unsigned long long)b * MAPW + (unsigned)r0) * (unsigned long long)(WW * 4));
    unsigned lds = (unsigned)(uintptr_t)&sF[r0 - gr0][0];  // LDS byte offset

    v4u g0;
    g0.x = 1u;                                  // count=1 (valid descriptor)
    g0.y = lds;                                 // lds_addr
    g0.z = (unsigned)ga;                        // global_addr[31:0]
    g0.w = (unsigned)(ga >> 32) | (2u << 30);   // global_addr[56:32] | type=2

    v8i_t g1;
    g1[0] = (int)(2u << 16);                            // data_size = 4 bytes
    g1[1] = (int)(((unsigned)WW   & 0xFFFFu) << 16);    // tensor_dim0 lo16
    g1[2] = (int)(((unsigned)MAPW & 0xFFFFu) << 16);    // dim0 hi16 | tensor_dim1 lo16
    g1[3] = (int)(((unsigned)WW) << 16);                // dim1 hi16 | tile_dim0=30
    g1[4] = (int)(unsigned)nrows;                       // tile_dim1 | tile_dim2=0
    g1[5] = (int)(unsigned)WW;                          // tensor_dim0_stride lo32
    g1[6] = 0;                                          // stride0 hi16 | stride1 lo16
    g1[7] = 0;                                          // stride1 hi32
    v4i_t g2 = {0, 0, 0, 0};                            // 2-D: groups 2/3 unused
    v4i_t g3 = {0, 0, 0, 0};
    v8i_t g4 = {0, 0, 0, 0, 0, 0, 0, 0};                // extra group (unused)

    __builtin_amdgcn_tensor_load_to_lds(g0, g1, g2, g3, g4, 0);
    __builtin_amdgcn_s_wait_tensorcnt(0);               // TENSORcnt == 0
  }
  __syncthreads();

  // --- Horizontal erosion, 7 half-width classes of disk(10):
  // |dy| : 0  1..4  5,6  7  8  9  10  ->  k = 10, 9, 8, 7, 6, 4, 0
  const int kList[7] = {0, 4, 6, 7, 8, 9, 10};
  for (int i = tid; i < 7 * ROWS * WW; i += 256) {
    int cls = i / (ROWS * WW);
    int rem = i % (ROWS * WW);
    int rr = rem / WW, w = rem % WW;
    unsigned M = sF[rr][w];
    unsigned L = (w > 0)      ? sF[rr][w - 1] : 0u;
    unsigned R = (w < WW - 1) ? sF[rr][w + 1] : 0u;
    unsigned res = M;
    int k = kList[cls];
    for (int d = 1; d <= k; ++d) {
      unsigned shr = (M >> d) | (R << (32 - d));   // pixel x+d
      unsigned shl = (M << d) | (L >> (32 - d));   // pixel x-d
      res &= shr & shl;
    }
    sE[cls][rr][w] = res;
  }
  __syncthreads();

  // --- Vertical AND across 21 rows (class chosen by |dy|): full disk erosion.
  const int clsOf[11] = {6, 5, 5, 5, 5, 4, 4, 3, 2, 1, 0};
  for (int i = tid; i < ERN * WW; i += 256) {
    int e = i / WW, w = i % WW;          // eroded global row = y0 - 1 + e
    int base = 10 + e;                   // sF/sE row of dy = 0
    unsigned res = 0xFFFFFFFFu;
    #pragma unroll
    for (int dy = -10; dy <= 10; ++dy) {
      int a = dy < 0 ? -dy : dy;
      res &= sE[clsOf[a]][base + dy][w];
    }
    sEr[e][w] = res;
  }
  __syncthreads();

  // --- Cross dilation (disk(1)) and border: goal = dilate(er) & ~er.
  for (int i = tid; i < BAND * WW; i += 256) {
    int y = i / WW, w = i % WW;          // output row y0 + y ; sEr row y+1
    unsigned C  = sEr[y + 1][w];
    unsigned U  = sEr[y][w];
    unsigned D  = sEr[y + 2][w];
    unsigned Lw = (w > 0)      ? sEr[y + 1][w - 1] : 0u;
    unsigned Rw = (w < WW - 1) ? sEr[y + 1][w + 1] : 0u;
    unsigned lef = (C << 1) | (Lw >> 31);   // pixel x-1
    unsigned rig = (C >> 1) | (Rw << 31);   // pixel x+1
    unsigned dil = C | U | D | lef | rig;
    sG[y][w] = dil & ~C;
  }
  __syncthreads();

  // --- Expand bits -> f32, coalesced stores.
  float* o = out + ((size_t)b * PLANE + (size_t)y0 * MAPW);
  for (int i = tid; i < BAND * MAPW; i += 256) {
    int y = i / MAPW, x = i - y * MAPW;
    unsigned bit = (sG[y][x >> 5] >> (x & 31)) & 1u;
    o[(size_t)y * MAPW + x] = (float)bit;
  }
}

// ---------------------------------------------------------------------------
extern "C" void kernel_launch(void* const* d_in, const int* in_sizes, int n_in,
                              void* d_out, int out_size, void* d_ws, size_t ws_size,
                              hipStream_t stream) {
  const float* mf  = (const float*)d_in[0];   // (8,24,960,960) f32
  float*       out = (float*)d_out;           // (8,1,960,960)  f32
  unsigned*    ws  = (unsigned*)d_ws;         // 8*960*30 u32 = 921,600 B

  dim3 pgrid((unsigned)(PLANE / 256u), 8);                        // (3600, 8)
  fep_pack<<<pgrid, 256, 0, stream>>>(mf, ws);

  dim3 grid(MAPW / BAND, 8);                                      // (60, 8)
  fep_morph<<<grid, 256, 0, stream>>>(ws, out);
}